// GMM_41626823033065
// MI455X (gfx1250) — compile-verified
//
#include <hip/hip_runtime.h>
#include <hip/hip_bf16.h>

typedef __attribute__((ext_vector_type(2))) float v2f;
typedef __attribute__((ext_vector_type(8))) float v8f;

#define DDIM 6
#define NFEAT 32   // 28 real features padded to 32 (8 WMMA k-steps of 4)

#define LOG2E 1.4426950408889634f
#define LN2   0.6931471805599453f

// ---------------------------------------------------------------------------
// Kernel 1: per-component preprocessing.
// Cholesky(Sigma) -> logdet, L^-1, P = L^-T L^-1, then emit a 32-float
// coefficient row theta_k (scaled by log2(e)) such that
//   phi(x_b) . theta_k = log2e * [ log w_k + log N(x_b; mu_k, Sigma_k) ]
// phi(x) = [1, x0..x5, x_i*x_j (i<=j, 21 terms), 0,0,0,0].
// Also writes alpha_k = log w_k - 0.5*(D log2pi + logdet) (natural log),
// an upper bound on the component's score (since quad >= 0).
// ---------------------------------------------------------------------------
__global__ void gmm_precompute(const float* __restrict__ means,
                               const float* __restrict__ covs,
                               const float* __restrict__ weights,
                               float* __restrict__ theta,
                               float* __restrict__ alphas, int K) {
    int k = blockIdx.x * blockDim.x + threadIdx.x;
    if (k >= K) return;

    float C[DDIM][DDIM];
#pragma unroll
    for (int i = 0; i < DDIM; ++i)
#pragma unroll
        for (int j = 0; j < DDIM; ++j)
            C[i][j] = covs[k * DDIM * DDIM + i * DDIM + j];

    // Cholesky: C = L L^T (lower)
    float L[DDIM][DDIM];
    float logdet = 0.0f;
#pragma unroll
    for (int j = 0; j < DDIM; ++j) {
        float s = C[j][j];
#pragma unroll
        for (int p = 0; p < DDIM; ++p)
            if (p < j) s -= L[j][p] * L[j][p];
        float d = sqrtf(s);
        L[j][j] = d;
        logdet += 2.0f * logf(d);
        float inv = 1.0f / d;
#pragma unroll
        for (int i = 0; i < DDIM; ++i)
            if (i > j) {
                float t = C[i][j];
#pragma unroll
                for (int p = 0; p < DDIM; ++p)
                    if (p < j) t -= L[i][p] * L[j][p];
                L[i][j] = t * inv;
            }
    }

    // Li = L^-1 (lower), forward substitution per column
    float Li[DDIM][DDIM];
#pragma unroll
    for (int j = 0; j < DDIM; ++j) {
        Li[j][j] = 1.0f / L[j][j];
#pragma unroll
        for (int i = 0; i < DDIM; ++i)
            if (i > j) {
                float s = 0.0f;
#pragma unroll
                for (int p = 0; p < DDIM; ++p)
                    if (p >= j && p < i) s += L[i][p] * Li[p][j];
                Li[i][j] = -s / L[i][i];
            }
    }

    // P = Li^T Li (precision matrix)
    float P[DDIM][DDIM];
#pragma unroll
    for (int a = 0; a < DDIM; ++a)
#pragma unroll
        for (int b = 0; b < DDIM; ++b)
            if (b >= a) {
                float s = 0.0f;
#pragma unroll
                for (int i = 0; i < DDIM; ++i)
                    if (i >= b) s += Li[i][a] * Li[i][b];
                P[a][b] = s;
                P[b][a] = s;
            }

    float mu[DDIM];
#pragma unroll
    for (int i = 0; i < DDIM; ++i) mu[i] = means[k * DDIM + i];

    float v[DDIM];
    float c = 0.0f;
#pragma unroll
    for (int a = 0; a < DDIM; ++a) {
        float s = 0.0f;
#pragma unroll
        for (int b = 0; b < DDIM; ++b) s += P[a][b] * mu[b];
        v[a] = s;
        c += mu[a] * s;
    }

    const float LOG2PI = 1.8378770664093453f;
    float alpha = logf(weights[k]) - 0.5f * ((float)DDIM * LOG2PI + logdet);
    alphas[k] = alpha;

    float th[NFEAT];
    th[0] = alpha - 0.5f * c;
#pragma unroll
    for (int i = 0; i < DDIM; ++i) th[1 + i] = v[i];
    int idx = 7;
#pragma unroll
    for (int i = 0; i < DDIM; ++i)
#pragma unroll
        for (int j = i; j < DDIM; ++j) {
            th[idx] = (i == j) ? (-0.5f * P[i][i]) : (-P[i][j]);
            ++idx;
        }
#pragma unroll
    for (int t = 28; t < NFEAT; ++t) th[t] = 0.0f;

#pragma unroll
    for (int t = 0; t < NFEAT; ++t) theta[k * NFEAT + t] = LOG2E * th[t];
}

// ---------------------------------------------------------------------------
// Kernel 1b: M = max_k alpha_k (global score upper bound); fold -log2e*M into
// each theta constant term so all GEMM outputs are <= 0 (exp2 never overflows).
// ---------------------------------------------------------------------------
__global__ void gmm_shift(float* __restrict__ theta,
                          const float* __restrict__ alphas,
                          float* __restrict__ Mout, int K) {
    __shared__ float red[256];
    int t = threadIdx.x;
    float mx = -1e30f;
    for (int k = t; k < K; k += 256) mx = fmaxf(mx, alphas[k]);
    red[t] = mx;
    __syncthreads();
    for (int off = 128; off > 0; off >>= 1) {
        if (t < off) red[t] = fmaxf(red[t], red[t + off]);
        __syncthreads();
    }
    float M = red[0];
    if (t == 0) *Mout = M;
    for (int k = t; k < K; k += 256) theta[k * NFEAT] -= LOG2E * M;
}

// ---------------------------------------------------------------------------
// Kernel 2: fused WMMA GEMM (Phi[B x 32] x Theta^T[32 x K]) + exp2-accumulate.
// One wave handles 32 rows (two 16-row C tiles sharing each B fragment);
// 8 waves per 256-thread block. Inner loop per 16-component tile:
//   8x global_load_b64 (theta) -> 16x v_wmma_f32_16x16x4_f32
//   -> 16x (v_exp_f32 + v_add) per lane.  No max bookkeeping needed.
// ---------------------------------------------------------------------------
__global__ __launch_bounds__(256) void gmm_logpdf(const float* __restrict__ x,
                                                  const float* __restrict__ theta,
                                                  const float* __restrict__ Mptr,
                                                  float* __restrict__ out,
                                                  int B, int K) {
    __shared__ float phi[8][32 * NFEAT];  // 4KB per wave, 32KB per block

    const int lane = threadIdx.x & 31;
    const int wave = threadIdx.x >> 5;
    const int row_base = (blockIdx.x * 8 + wave) * 32;
    float* ph = &phi[wave][0];

    // --- build phi: all 32 lanes, one row each ---
    {
        int r = row_base + lane;
        if (r > B - 1) r = B - 1;  // clamp (B divisible by 32 in practice)
        float xv[DDIM];
#pragma unroll
        for (int i = 0; i < DDIM; ++i) xv[i] = x[r * DDIM + i];
        float f[NFEAT];
        f[0] = 1.0f;
#pragma unroll
        for (int i = 0; i < DDIM; ++i) f[1 + i] = xv[i];
        int idx = 7;
#pragma unroll
        for (int i = 0; i < DDIM; ++i)
#pragma unroll
            for (int j = i; j < DDIM; ++j) {
                f[idx] = xv[i] * xv[j];
                ++idx;
            }
#pragma unroll
        for (int t = 28; t < NFEAT; ++t) f[t] = 0.0f;
#pragma unroll
        for (int t = 0; t < NFEAT; ++t) ph[lane * NFEAT + t] = f[t];
    }
    __syncthreads();

    // --- A fragments for both row tiles (constant across the k loop) ---
    // f32 16x16x4 A layout: lanes 0-15 rows 0-15 K=4t..4t+1; lanes 16-31 K=4t+2..4t+3
    const int arow = lane & 15;
    const int acol = (lane >> 4) * 2;
    v2f a0[8], a1[8];
#pragma unroll
    for (int t = 0; t < 8; ++t) {
        a0[t] = *(const v2f*)(ph + arow * NFEAT + 4 * t + acol);
        a1[t] = *(const v2f*)(ph + (arow + 16) * NFEAT + 4 * t + acol);
    }

    float s0[8], s1[8];
#pragma unroll
    for (int i = 0; i < 8; ++i) { s0[i] = 0.0f; s1[i] = 0.0f; }

    const int comp = lane & 15;
    const int bcol = (lane >> 4) * 2;
    const int ntiles = K >> 4;

    for (int nt = 0; nt < ntiles; ++nt) {
        const float* trow = theta + (size_t)(nt * 16 + comp) * NFEAT;
        v2f b[8];
#pragma unroll
        for (int t = 0; t < 8; ++t)
            b[t] = *(const v2f*)(trow + 4 * t + bcol);

        v8f c0 = {};
        v8f c1 = {};
#pragma unroll
        for (int t = 0; t < 8; ++t) {
            c0 = __builtin_amdgcn_wmma_f32_16x16x4_f32(
                false, a0[t], false, b[t], (short)0, c0, false, false);
            c1 = __builtin_amdgcn_wmma_f32_16x16x4_f32(
                false, a1[t], false, b[t], (short)0, c1, false, false);
        }
        // scores are log2-domain and <= 0: just exp2 + accumulate
#pragma unroll
        for (int i = 0; i < 8; ++i) {
            s0[i] += __builtin_amdgcn_exp2f(c0[i]);
            s1[i] += __builtin_amdgcn_exp2f(c1[i]);
        }
    }

    // --- sum across the 16 columns (xor shuffles stay within 16-lane halves) ---
#pragma unroll
    for (int i = 0; i < 8; ++i) {
#pragma unroll
        for (int mask = 1; mask < 16; mask <<= 1) {
            s0[i] += __shfl_xor(s0[i], mask, 32);
            s1[i] += __shfl_xor(s1[i], mask, 32);
        }
    }

    const float Mv = *Mptr;
    // lanes 0 / 16 hold rows 0-7 / 8-15 of each tile
    if ((lane & 15) == 0) {
        int rb = row_base + (lane >> 4) * 8;
#pragma unroll
        for (int i = 0; i < 8; ++i)
            if (rb + i < B) out[rb + i] = Mv + LN2 * __builtin_amdgcn_logf(s0[i]);
        rb += 16;
#pragma unroll
        for (int i = 0; i < 8; ++i)
            if (rb + i < B) out[rb + i] = Mv + LN2 * __builtin_amdgcn_logf(s1[i]);
    }
}

// ---------------------------------------------------------------------------
extern "C" void kernel_launch(void* const* d_in, const int* in_sizes, int n_in,
                              void* d_out, int out_size, void* d_ws, size_t ws_size,
                              hipStream_t stream) {
    const float* x       = (const float*)d_in[0];
    const float* means   = (const float*)d_in[1];
    const float* covs    = (const float*)d_in[2];
    const float* weights = (const float*)d_in[3];
    float* out = (float*)d_out;

    const int B = in_sizes[0] / DDIM;   // 16384
    const int K = in_sizes[3];          // 1024

    float* theta  = (float*)d_ws;           // K*32 floats
    float* alphas = theta + (size_t)K * NFEAT;  // K floats
    float* Mout   = alphas + K;             // 1 float

    int pre_blocks = (K + 255) / 256;
    gmm_precompute<<<pre_blocks, 256, 0, stream>>>(means, covs, weights, theta, alphas, K);
    gmm_shift<<<1, 256, 0, stream>>>(theta, alphas, Mout, K);

    int row_tiles = (B + 31) / 32;          // 32 rows per wave
    int blocks = (row_tiles + 7) / 8;       // 8 waves per block
    gmm_logpdf<<<blocks, 256, 0, stream>>>(x, theta, Mout, out, B, K);
}